// ConformanceGNN_28458453303307
// MI455X (gfx1250) — compile-verified
//
#include <hip/hip_runtime.h>
#include <hip/hip_bf16.h>
#include <math.h>

typedef __attribute__((ext_vector_type(2))) float v2f;
typedef __attribute__((ext_vector_type(8))) float v8f;

#define HD 128

// ============================================================================
// GEMM: out[N,128] = A[N,K] @ W[128,K]^T + bias   (optionally + res, relu)
// One 256-thread block = 8 waves; block owns 16 rows, wave w owns cols [16w,16w+16).
// Uses V_WMMA_F32_16X16X4_F32, K-loop in steps of 4.
// A tile staged in LDS with stride K+2 (bank-conflict padding).
// If SPLIT: logical A = concat(A0[N,128], A1[N,128]) along columns (K must be 256).
// Full-tile fast path: when m0+16 <= N (always true for N%16==0) all bounds
// checks vanish -> straight-line store clause instead of 8 divergent branches.
// ============================================================================
template<int K, bool SPLIT, bool RESRELU>
__global__ __launch_bounds__(256)
void gemm128_kernel(const float* A0, const float* A1,
                    const float* __restrict__ W, const float* __restrict__ bias,
                    const float* res, float* out, int N)
{
    constexpr int LS = K + 2;
    __shared__ float As[16 * LS];
    const int m0  = blockIdx.x * 16;
    const int tid = threadIdx.x;
    const bool full = (m0 + 16 <= N);   // block-uniform

    if (full) {
        for (int i = tid; i < 16 * K; i += 256) {
            int r = i / K, c = i - r * K;
            int row = m0 + r;
            float v;
            if (SPLIT) v = (c < HD) ? A0[row * HD + c] : A1[row * HD + (c - HD)];
            else       v = A0[row * K + c];
            As[r * LS + c] = v;
        }
    } else {
        for (int i = tid; i < 16 * K; i += 256) {
            int r = i / K, c = i - r * K;
            int row = m0 + r;
            float v = 0.0f;
            if (row < N) {
                if (SPLIT) v = (c < HD) ? A0[row * HD + c] : A1[row * HD + (c - HD)];
                else       v = A0[row * K + c];
            }
            As[r * LS + c] = v;
        }
    }
    __syncthreads();

    const int wave = tid >> 5;
    const int lane = tid & 31;
    const int half = lane >> 4;   // 0: lanes 0-15, 1: lanes 16-31
    const int lr   = lane & 15;
    const int n0   = wave * 16;

    v8f acc = {};
    #pragma unroll 4
    for (int k = 0; k < K; k += 4) {
        const int ka = k + 2 * half;
        v2f a, b;
        // A 16x4 fragment: lanes 0-15 hold K=k..k+1, lanes 16-31 hold K=k+2..k+3
        a.x = As[lr * LS + ka];
        a.y = As[lr * LS + ka + 1];
        // B 4x16 fragment: B[k][n] = W[n][k] (W is [out,in] row-major)
        const float* wr = W + (n0 + lr) * K + ka;
        b.x = wr[0];
        b.y = wr[1];
        acc = __builtin_amdgcn_wmma_f32_16x16x4_f32(false, a, false, b,
                                                    (short)0, acc, false, false);
    }

    const float bv = bias[n0 + lr];
    if (full) {
        #pragma unroll
        for (int r = 0; r < 8; ++r) {
            int row = m0 + r + 8 * half;   // C/D: VGPR r -> row r (lanes 0-15) / r+8
            int o = row * HD + n0 + lr;
            float v = acc[r] + bv;
            if (RESRELU) v = fmaxf(v + res[o], 0.0f);
            out[o] = v;
        }
    } else {
        #pragma unroll
        for (int r = 0; r < 8; ++r) {
            int row = m0 + r + 8 * half;
            if (row < N) {
                int o = row * HD + n0 + lr;
                float v = acc[r] + bv;
                if (RESRELU) v = fmaxf(v + res[o], 0.0f);
                out[o] = v;
            }
        }
    }
}

// ---------------------------------------------------------------------------
__global__ __launch_bounds__(256)
void zero_kernel(float* p, int n) {
    int i = blockIdx.x * 256 + threadIdx.x;
    if (i < n) p[i] = 0.0f;
}

__global__ __launch_bounds__(256)
void degree_kernel(const int* __restrict__ idx, float* deg, int E) {
    int e = blockIdx.x * 256 + threadIdx.x;
    if (e < E) atomicAdd(&deg[idx[e]], 1.0f);
}

// logit[v] = mnode[v] . w_att + b_att        (one wave per row)
__global__ __launch_bounds__(256)
void att_logit_kernel(const float* __restrict__ m, const float* __restrict__ w,
                      const float* __restrict__ b, float* __restrict__ logit, int N)
{
    int wid  = (blockIdx.x * 256 + threadIdx.x) >> 5;
    int lane = threadIdx.x & 31;
    int nw   = (gridDim.x * 256) >> 5;
    for (int row = wid; row < N; row += nw) {
        const float* mr = m + row * HD;
        float s = 0.0f;
        #pragma unroll
        for (int c = lane; c < HD; c += 32) s += mr[c] * w[c];
        #pragma unroll
        for (int off = 16; off > 0; off >>= 1) s += __shfl_down(s, off, 32);
        if (lane == 0) logit[row] = s + b[0];
    }
}

// stats[0] = max_v logit[v];  stats[1] = sum_v deg[v]*exp(logit[v]-max)
__global__ __launch_bounds__(1024)
void softmax_stats_kernel(const float* __restrict__ logit, const float* __restrict__ deg,
                          int N, float* __restrict__ stats)
{
    __shared__ float red[1024];
    int t = threadIdx.x;
    float mx = -3.0e38f;
    for (int i = t; i < N; i += 1024) mx = fmaxf(mx, logit[i]);
    red[t] = mx; __syncthreads();
    for (int s = 512; s > 0; s >>= 1) {
        if (t < s) red[t] = fmaxf(red[t], red[t + s]);
        __syncthreads();
    }
    float m = red[0];
    __syncthreads();
    float sum = 0.0f;
    for (int i = t; i < N; i += 1024) sum += deg[i] * __expf(logit[i] - m);
    red[t] = sum; __syncthreads();
    for (int s = 512; s > 0; s >>= 1) {
        if (t < s) red[t] += red[t + s];
        __syncthreads();
    }
    if (t == 0) { stats[0] = m; stats[1] = red[0]; }
}

// mnode[v,:] *= exp(logit[v]-max)/Z     (in place)
__global__ __launch_bounds__(256)
void scale_kernel(float* m, const float* __restrict__ logit,
                  const float* __restrict__ stats, int N)
{
    int i = blockIdx.x * 256 + threadIdx.x;
    if (i < N * HD) {
        int row = i >> 7;
        m[i] *= __expf(logit[row] - stats[0]) / stats[1];
    }
}

// msgs[dst[e],:] += scaled[src[e],:]   (thread = one (edge, channel))
__global__ __launch_bounds__(256)
void scatter_kernel(const int* __restrict__ src, const int* __restrict__ dst,
                    const float* __restrict__ val, float* out, int E)
{
    int gid = blockIdx.x * 256 + threadIdx.x;
    int e = gid >> 7, c = gid & 127;
    if (e < E) atomicAdd(&out[dst[e] * HD + c], val[src[e] * HD + c]);
}

// column-sum of h[N,128] into sum[128] (sum pre-zeroed)
__global__ __launch_bounds__(256)
void colsum_kernel(const float* __restrict__ h, int N, float* sum)
{
    __shared__ float ls[HD];
    int t = threadIdx.x;
    if (t < HD) ls[t] = 0.0f;
    __syncthreads();
    int c = t & 127;
    float acc = 0.0f;
    for (int row = blockIdx.x * 2 + (t >> 7); row < N; row += gridDim.x * 2)
        acc += h[row * HD + c];
    atomicAdd(&ls[c], acc);
    __syncthreads();
    if (t < HD) atomicAdd(&sum[c], ls[c]);
}

// pg = (sumP/P) @ Wp^T + bp ;  tg = (sumT/T) @ Wt^T + bt
__global__ __launch_bounds__(256)
void pool_kernel(const float* __restrict__ sumP, const float* __restrict__ sumT,
                 const float* __restrict__ Wp, const float* __restrict__ bp,
                 const float* __restrict__ Wt, const float* __restrict__ bt,
                 float invP, float invT, float* pg, float* tg)
{
    int t = threadIdx.x;
    const float* s = (t < HD) ? sumP : sumT;
    const float* W = (t < HD) ? Wp : Wt;
    const float* b = (t < HD) ? bp : bt;
    float inv      = (t < HD) ? invP : invT;
    int c = t & 127;
    float acc = 0.0f;
    for (int k = 0; k < HD; ++k) acc += W[c * HD + k] * (s[k] * inv);
    if (t < HD) pg[c] = acc + b[c];
    else        tg[c] = acc + b[c];
}

// prefix_h = prefix[64] @ W^T + b   (W is [128,64])
__global__ __launch_bounds__(128)
void prefix_kernel(const float* __restrict__ x, const float* __restrict__ W,
                   const float* __restrict__ b, float* out)
{
    int c = threadIdx.x;
    float acc = 0.0f;
    for (int k = 0; k < 64; ++k) acc += W[c * 64 + k] * x[k];
    out[c] = acc + b[c];
}

// both 3-layer MLP heads on combined[384]; out[0..127]=pred sigmoid, out[128]=clf sigmoid
__global__ __launch_bounds__(256)
void head_kernel(const float* __restrict__ pg, const float* __restrict__ tg,
                 const float* __restrict__ px,
                 const float* __restrict__ w0, const float* __restrict__ b0,
                 const float* __restrict__ w1, const float* __restrict__ b1,
                 const float* __restrict__ w2, const float* __restrict__ b2,
                 const float* __restrict__ c0w, const float* __restrict__ c0b,
                 const float* __restrict__ c1w, const float* __restrict__ c1b,
                 const float* __restrict__ c2w, const float* __restrict__ c2b,
                 float* out)
{
    __shared__ float comb[384];
    __shared__ float h1[256];
    __shared__ float h2[128];
    int t = threadIdx.x;
    if (t < 128) { comb[t] = pg[t]; comb[128 + t] = tg[t]; comb[256 + t] = px[t]; }
    __syncthreads();
    // predictor
    { float a = b0[t]; for (int k = 0; k < 384; ++k) a += w0[t * 384 + k] * comb[k];
      h1[t] = fmaxf(a, 0.0f); }
    __syncthreads();
    if (t < 128) { float a = b1[t]; for (int k = 0; k < 256; ++k) a += w1[t * 256 + k] * h1[k];
                   h2[t] = fmaxf(a, 0.0f); }
    __syncthreads();
    if (t < 128) { float a = b2[t]; for (int k = 0; k < 128; ++k) a += w2[t * 128 + k] * h2[k];
                   out[t] = 1.0f / (1.0f + __expf(-a)); }
    __syncthreads();
    // classifier
    { float a = c0b[t]; for (int k = 0; k < 384; ++k) a += c0w[t * 384 + k] * comb[k];
      h1[t] = fmaxf(a, 0.0f); }
    __syncthreads();
    if (t < 128) { float a = c1b[t]; for (int k = 0; k < 256; ++k) a += c1w[t * 256 + k] * h1[k];
                   h2[t] = fmaxf(a, 0.0f); }
    __syncthreads();
    if (t == 0) { float a = c2b[0]; for (int k = 0; k < 128; ++k) a += c2w[k] * h2[k];
                  out[128] = 1.0f / (1.0f + __expf(-a)); }
}

// ============================================================================
// Param leaf order (JAX pytree: dict keys sorted, "b" < "w"):
//  0..4 : place_features, transition_features, prefix_encoding, pre_ei, post_ei
//  5..10: clf  [b0,w0,b1,w1,b2,w2]
// 11+12l: layer l: p2t.b,p2t.w, p_att.b,p_att.w, p_upd.b,p_upd.w,
//                  t2p.b,t2p.w, t_att.b,t_att.w, t_upd.b,t_upd.w
// 47,48 : place_emb b,w     49,50: place_pool b,w
// 51..56: pred [b0,w0,b1,w1,b2,w2]
// 57,58 : prefix_emb b,w    59,60: trans_emb b,w    61,62: trans_pool b,w
// ============================================================================
extern "C" void kernel_launch(void* const* d_in, const int* in_sizes, int n_in,
                              void* d_out, int out_size, void* d_ws, size_t ws_size,
                              hipStream_t stream)
{
    const float* place_feat = (const float*)d_in[0];
    const float* trans_feat = (const float*)d_in[1];
    const float* prefix     = (const float*)d_in[2];
    const int*   pre_src    = (const int*)d_in[3];
    const int*   post_src   = (const int*)d_in[4];

    const int P = in_sizes[0] / 16;
    const int T = in_sizes[1] / 16;
    const int E = in_sizes[3] / 2;
    const int* pre_dst  = pre_src  + E;
    const int* post_dst = post_src + E;

    auto F = [&](int i) { return (const float*)d_in[i]; };

    // ---- workspace layout (floats) ----
    const int NB = ((P > T) ? P : T) * HD;      // 6.4M floats per node buffer
    float* ws    = (float*)d_ws;
    float* ph    = ws;                // [P,128]
    float* th    = ws + (size_t)NB;   // [T,128]
    float* mnode = ws + 2 * (size_t)NB;
    float* msgsT = ws + 3 * (size_t)NB;
    float* msgsP = ws + 4 * (size_t)NB;
    float* logit = ws + 5 * (size_t)NB;         // max(P,T)
    float* degP  = logit + 65536;
    float* degT  = degP + 65536;
    float* sm    = degT + 65536;                // small scalars/vectors
    float* stats    = sm + 0;    // 2
    float* prefix_h = sm + 32;   // 128
    float* csum_p   = sm + 192;  // 128
    float* csum_t   = sm + 352;  // 128
    float* pg       = sm + 512;  // 128
    float* tg       = sm + 672;  // 128

    const int gP = (P + 15) / 16, gT = (T + 15) / 16;
    const int gPH = (P * HD + 255) / 256, gTH = (T * HD + 255) / 256;
    const int gE  = (E + 255) / 256, gEH = (E * HD + 255) / 256;

    // zero degrees + smalls region (contiguous: degP, degT, sm[1024])
    zero_kernel<<<(2 * 65536 + 1024 + 255) / 256, 256, 0, stream>>>(degP, 2 * 65536 + 1024);

    // source out-degrees (for global-softmax denominator)
    degree_kernel<<<gE, 256, 0, stream>>>(pre_src,  degP, E);
    degree_kernel<<<gE, 256, 0, stream>>>(post_src, degT, E);

    // embeddings: [N,16] -> [N,128]
    gemm128_kernel<16, false, false><<<gP, 256, 0, stream>>>(place_feat, nullptr, F(48), F(47), nullptr, ph, P);
    gemm128_kernel<16, false, false><<<gT, 256, 0, stream>>>(trans_feat, nullptr, F(60), F(59), nullptr, th, T);
    prefix_kernel<<<1, 128, 0, stream>>>(prefix, F(58), F(57), prefix_h);

    for (int l = 0; l < 3; ++l) {
        const int b = 11 + l * 12;
        // ---- place -> transition (uses t_att) ----
        gemm128_kernel<128, false, false><<<gP, 256, 0, stream>>>(ph, nullptr, F(b + 1), F(b + 0), nullptr, mnode, P);
        att_logit_kernel<<<512, 256, 0, stream>>>(mnode, F(b + 9), F(b + 8), logit, P);
        softmax_stats_kernel<<<1, 1024, 0, stream>>>(logit, degP, P, stats);
        scale_kernel<<<gPH, 256, 0, stream>>>(mnode, logit, stats, P);
        zero_kernel<<<gTH, 256, 0, stream>>>(msgsT, T * HD);
        scatter_kernel<<<gEH, 256, 0, stream>>>(pre_src, pre_dst, mnode, msgsT, E);
        // ---- transition -> place (uses p_att) ----
        gemm128_kernel<128, false, false><<<gT, 256, 0, stream>>>(th, nullptr, F(b + 7), F(b + 6), nullptr, mnode, T);
        att_logit_kernel<<<512, 256, 0, stream>>>(mnode, F(b + 3), F(b + 2), logit, T);
        softmax_stats_kernel<<<1, 1024, 0, stream>>>(logit, degT, T, stats);
        scale_kernel<<<gTH, 256, 0, stream>>>(mnode, logit, stats, T);
        zero_kernel<<<gPH, 256, 0, stream>>>(msgsP, P * HD);
        scatter_kernel<<<gEH, 256, 0, stream>>>(post_src, post_dst, mnode, msgsP, E);
        // ---- residual updates (in place; each block owns disjoint rows) ----
        gemm128_kernel<256, true, true><<<gP, 256, 0, stream>>>(ph, msgsP, F(b + 5), F(b + 4),  ph, ph, P);
        gemm128_kernel<256, true, true><<<gT, 256, 0, stream>>>(th, msgsT, F(b + 11), F(b + 10), th, th, T);
    }

    // pooling: mean commutes with affine pool layer
    colsum_kernel<<<256, 256, 0, stream>>>(ph, P, csum_p);
    colsum_kernel<<<256, 256, 0, stream>>>(th, T, csum_t);
    pool_kernel<<<1, 256, 0, stream>>>(csum_p, csum_t, F(50), F(49), F(62), F(61),
                                       1.0f / (float)P, 1.0f / (float)T, pg, tg);

    head_kernel<<<1, 256, 0, stream>>>(pg, tg, prefix_h,
                                       F(52), F(51), F(54), F(53), F(56), F(55),
                                       F(6),  F(5),  F(8),  F(7),  F(10), F(9),
                                       (float*)d_out);
}